// Classify1_32444182954704
// MI455X (gfx1250) — compile-verified
//
#include <hip/hip_runtime.h>
#include <math.h>

typedef float v2f __attribute__((ext_vector_type(2)));
typedef float v8f __attribute__((ext_vector_type(8)));

#define K_TOP 20
#define N_PTS 8192
#define M_PTS 8192
#define NB    2
#define BN_EPS 1e-5f

// ---------------------------------------------------------------------------
// Phase 1: fused pairwise-distance + per-row top-20.
//   pd[n,m] = 2*<x_n,y_m> - xx_n - yy_m
// One V_WMMA_F32_16X16X4_F32 per 16x16 tile computes  2*inner - yy  directly:
//   A (16x4) = [y0, y1, y2, yy]  (M = y-point index)
//   B (4x16) = [2*x0, 2*x1, 2*x2, -1]  (N = x-point index)
// D layout (32-bit C/D 16x16): lane L holds column N = L%16; VGPR j holds
// row M = j + 8*(L/16).  So each lane owns one query point n and sees 8
// candidate m's per tile -> register-resident sorted top-20 per lane.
// 4 waves split the m-range per n-tile; 8 sorted lists merged via LDS.
// Block = 256 threads = 8 waves = 2 n-tiles per block.
// ---------------------------------------------------------------------------
__global__ __launch_bounds__(256)
void knn_topk_kernel(const float* __restrict__ x,   // [B,N,3]
                     const float* __restrict__ y,   // [B,M,3]
                     float* __restrict__ dist)      // [B,N,20]
{
    __shared__ float lds[2][4][32][K_TOP];

    const int wave = threadIdx.x >> 5;
    const int lane = threadIdx.x & 31;
    const int half = lane >> 4;      // 0: lanes 0-15, 1: lanes 16-31
    const int l16  = lane & 15;
    const int t    = wave >> 2;      // n-tile slot within block (0..1)
    const int wq   = wave & 3;       // m-chunk index (0..3)

    const int tile = blockIdx.x * 2 + t;            // global n-tile id
    const int b    = tile / (N_PTS / 16);
    const int n0   = (tile % (N_PTS / 16)) * 16;
    const int n    = n0 + l16;

    // Resident B fragment: x tile (4x16), rows scaled by 2, row 3 = -1.
    const float* xp = x + ((size_t)b * N_PTS + n) * 3;
    const float x0 = xp[0], x1 = xp[1], x2 = xp[2];
    const float xx = x0 * x0 + x1 * x1 + x2 * x2;
    v2f bfrag;
    if (half == 0) { bfrag.x = 2.f * x0; bfrag.y = 2.f * x1; }
    else           { bfrag.x = 2.f * x2; bfrag.y = -1.f;     }

    float best[K_TOP];                // sorted descending, static indices only
    #pragma unroll
    for (int i = 0; i < K_TOP; ++i) best[i] = -3.0e38f;

    const float* yb = y + (size_t)b * M_PTS * 3;
    const int mbeg = wq * (M_PTS / 4);
    const int mend = mbeg + (M_PTS / 4);

    for (int m0 = mbeg; m0 < mend; m0 += 16) {
        // A fragment: y tile (16x4), column 3 holds yy.
        const float* yp = yb + (size_t)(m0 + l16) * 3;
        const float y0 = yp[0], y1 = yp[1], y2 = yp[2];
        const float yy = y0 * y0 + y1 * y1 + y2 * y2;
        v2f afrag;
        if (half == 0) { afrag.x = y0; afrag.y = y1; }
        else           { afrag.x = y2; afrag.y = yy; }

        v8f acc = {0.f, 0.f, 0.f, 0.f, 0.f, 0.f, 0.f, 0.f};
        acc = __builtin_amdgcn_wmma_f32_16x16x4_f32(
                  false, afrag, false, bfrag, (short)0, acc, false, false);

        // acc[j] = 2*<x_n, y_m> - yy_m   for m = m0 + j + 8*half
        #pragma unroll
        for (int j = 0; j < 8; ++j) {
            const float c = acc[j];
            if (c > best[K_TOP - 1]) {          // rare after warm-up
                float cur = c;
                #pragma unroll
                for (int i = 0; i < K_TOP; ++i) {   // static-index insertion
                    const float hi = fmaxf(best[i], cur);
                    cur     = fminf(best[i], cur);
                    best[i] = hi;
                }
            }
        }
    }

    #pragma unroll
    for (int i = 0; i < K_TOP; ++i) lds[t][wq][lane][i] = best[i];
    __syncthreads();

    // Merge the 8 sorted lists per query point; wave {4t}, lanes 0-15 do it.
    if (wq == 0 && half == 0) {
        int p[8];
        #pragma unroll
        for (int s = 0; s < 8; ++s) p[s] = 0;
        float* out = dist + ((size_t)b * N_PTS + n) * K_TOP;
        for (int i = 0; i < K_TOP; ++i) {
            float vmax = -3.4e38f; int smax = 0;
            #pragma unroll
            for (int s = 0; s < 8; ++s) {
                const float v = lds[t][s >> 1][(s & 1) * 16 + l16][p[s]];
                if (v > vmax) { vmax = v; smax = s; }
            }
            #pragma unroll
            for (int s = 0; s < 8; ++s) p[s] += (smax == s) ? 1 : 0;
            out[i] = vmax - xx;          // apply the per-row constant here
        }
    }
}

// ---------------------------------------------------------------------------
// Phase 2: per-point MLP 20 -> 256 -> 128 -> 1 with folded BN, via WMMA f32.
// Block = 256 threads = 8 waves, one 16-point column tile per block.
// GEMM1: M=256 (wave does row tiles w and w+8), K=20 = 5 WMMA k-steps.
// GEMM2: M=128 (wave does row tile w),          K=256 = 64 WMMA k-steps.
// GEMM3: 1x128 scalar dot + sigmoid.
// ---------------------------------------------------------------------------
__global__ __launch_bounds__(256)
void mlp_kernel(const float* __restrict__ dist,   // [B,N,20]
                const float* __restrict__ W1,     // [256,20]
                const float* __restrict__ g1, const float* __restrict__ be1,
                const float* __restrict__ mu1, const float* __restrict__ va1,
                const float* __restrict__ W2,     // [128,256]
                const float* __restrict__ g2, const float* __restrict__ be2,
                const float* __restrict__ mu2, const float* __restrict__ va2,
                const float* __restrict__ W3,     // [128]
                float* __restrict__ out)          // [B,N]
{
    __shared__ float h1s[256][16];
    __shared__ float h2s[128][16];

    const int wave = threadIdx.x >> 5;
    const int lane = threadIdx.x & 31;
    const int half = lane >> 4;
    const int l16  = lane & 15;

    const int tile = blockIdx.x;
    const int b    = tile / (N_PTS / 16);
    const int n0   = (tile % (N_PTS / 16)) * 16;

    // B fragments for GEMM1 (feature tile, K=20 -> 5 steps), reused per wave.
    const float* frow = dist + ((size_t)b * N_PTS + n0 + l16) * K_TOP;
    v2f bf[5];
    #pragma unroll
    for (int c = 0; c < 5; ++c) {
        const float2 f2 = *(const float2*)(frow + 4 * c + 2 * half);
        bf[c].x = f2.x; bf[c].y = f2.y;
    }

    // ---- GEMM1 + BN1 + ReLU -> h1s -------------------------------------
    #pragma unroll
    for (int rr = 0; rr < 2; ++rr) {
        const int r = wave + 8 * rr;          // row tile (0..15)
        const int m = 16 * r + l16;           // A-matrix row for this lane
        v8f acc = {0.f, 0.f, 0.f, 0.f, 0.f, 0.f, 0.f, 0.f};
        #pragma unroll
        for (int c = 0; c < 5; ++c) {
            const float2 w = *(const float2*)(W1 + (size_t)m * 20 + 4 * c + 2 * half);
            v2f af; af.x = w.x; af.y = w.y;
            acc = __builtin_amdgcn_wmma_f32_16x16x4_f32(
                      false, af, false, bf[c], (short)0, acc, false, false);
        }
        #pragma unroll
        for (int j = 0; j < 8; ++j) {
            const int mo = 16 * r + j + 8 * half;
            const float s = g1[mo] * rsqrtf(va1[mo] + BN_EPS);
            const float tt = be1[mo] - mu1[mo] * s;
            h1s[mo][l16] = fmaxf(s * acc[j] + tt, 0.f);
        }
    }
    __syncthreads();

    // ---- GEMM2 + BN2 + ReLU -> h2s -------------------------------------
    {
        const int m = 16 * wave + l16;
        v8f acc = {0.f, 0.f, 0.f, 0.f, 0.f, 0.f, 0.f, 0.f};
        for (int c = 0; c < 64; ++c) {
            const float2 w = *(const float2*)(W2 + (size_t)m * 256 + 4 * c + 2 * half);
            v2f af;  af.x = w.x;  af.y = w.y;
            v2f bf2;
            bf2.x = h1s[4 * c + 2 * half + 0][l16];
            bf2.y = h1s[4 * c + 2 * half + 1][l16];
            acc = __builtin_amdgcn_wmma_f32_16x16x4_f32(
                      false, af, false, bf2, (short)0, acc, false, false);
        }
        #pragma unroll
        for (int j = 0; j < 8; ++j) {
            const int mo = 16 * wave + j + 8 * half;
            const float s = g2[mo] * rsqrtf(va2[mo] + BN_EPS);
            const float tt = be2[mo] - mu2[mo] * s;
            h2s[mo][l16] = fmaxf(s * acc[j] + tt, 0.f);
        }
    }
    __syncthreads();

    // ---- GEMM3 (1x128) + sigmoid ----------------------------------------
    if (threadIdx.x < 16) {
        float a = 0.f;
        for (int c = 0; c < 128; ++c) a += W3[c] * h2s[c][threadIdx.x];
        out[(size_t)b * N_PTS + n0 + threadIdx.x] = 1.0f / (1.0f + expf(-a));
    }
}

// ---------------------------------------------------------------------------
extern "C" void kernel_launch(void* const* d_in, const int* in_sizes, int n_in,
                              void* d_out, int out_size, void* d_ws, size_t ws_size,
                              hipStream_t stream) {
    const float* x   = (const float*)d_in[0];
    const float* y   = (const float*)d_in[1];
    const float* W1  = (const float*)d_in[2];
    const float* g1  = (const float*)d_in[3];
    const float* be1 = (const float*)d_in[4];
    const float* mu1 = (const float*)d_in[5];
    const float* va1 = (const float*)d_in[6];
    const float* W2  = (const float*)d_in[7];
    const float* g2  = (const float*)d_in[8];
    const float* be2 = (const float*)d_in[9];
    const float* mu2 = (const float*)d_in[10];
    const float* va2 = (const float*)d_in[11];
    const float* W3  = (const float*)d_in[12];
    // d_in[13] = k (==20, baked in as K_TOP)

    float* dist = (float*)d_ws;                 // [B,N,20]  (1.31 MB)
    float* outp = (float*)d_out;                // [B,N,1]

    const int n_tiles = NB * (N_PTS / 16);      // 1024
    knn_topk_kernel<<<n_tiles / 2, 256, 0, stream>>>(x, y, dist);
    mlp_kernel<<<n_tiles, 256, 0, stream>>>(dist, W1, g1, be1, mu1, va1,
                                            W2, g2, be2, mu2, va2, W3, outp);
}